// Cross_Domain_Class_Alignment_27848567947850
// MI455X (gfx1250) — compile-verified
//
#include <hip/hip_runtime.h>

typedef __attribute__((ext_vector_type(2))) float v2f;
typedef __attribute__((ext_vector_type(8))) float v8f;

#define Bsz   8
#define Cdim  256
#define Hh    65
#define Ww    129
#define HW    (Hh * Ww)        // 8385
#define NPIX  (Bsz * HW)       // 67080
#define Kcen  19
#define OUTH  512
#define OUTW  1024
#define PERT  (Bsz * OUTH * OUTW)  // 4194304 per tensor
#define LDS_STRIDE 260         // 256 + 4 pad floats -> row stride 1040B, avoids 64-bank conflicts
#define WAVES 8
#define PIX_PER_BLOCK (WAVES * 16)

// Stage 1: fused GEMM (fp32 WMMA 16x16x4) + argmin over centroids.
// grid.y = 0: feature_s2t vs centroid_target ; grid.y = 1: feature_target vs centroid_s2t
__global__ __launch_bounds__(256) void cls_argmin_kernel(
    const float* __restrict__ feat_s2t, const float* __restrict__ feat_tgt,
    const float* __restrict__ cent_s2t, const float* __restrict__ cent_tgt,
    int* __restrict__ cls_out)
{
    __shared__ float lds_cent[Kcen * LDS_STRIDE];
    __shared__ float lds_csq[Kcen];

    const int t = blockIdx.y;
    const float* __restrict__ feat = (t == 0) ? feat_s2t : feat_tgt;
    const float* __restrict__ cent = (t == 0) ? cent_tgt : cent_s2t;  // cross-domain pairing

    // Preload centroids [19,256] into LDS (padded rows).
    for (int i = threadIdx.x; i < Kcen * Cdim; i += 256) {
        int r = i >> 8, c = i & 255;
        lds_cent[r * LDS_STRIDE + c] = cent[i];
    }
    __syncthreads();
    if (threadIdx.x < Kcen) {
        float s = 0.0f;
        for (int c = 0; c < Cdim; ++c) {
            float v = lds_cent[threadIdx.x * LDS_STRIDE + c];
            s += v * v;
        }
        lds_csq[threadIdx.x] = s;
    }
    __syncthreads();

    const int lane = threadIdx.x & 31;
    const int wave = threadIdx.x >> 5;
    const int tile = (blockIdx.x * WAVES + wave) * 16;   // first pixel of this wave's tile

    // Per-lane A-fragment address setup (16x4 f32 layout: M = lane&15, klo = 0 or 2).
    int p = tile + (lane & 15);
    if (p >= NPIX) p = NPIX - 1;                 // clamp: keep EXEC all-ones for WMMA
    const int b = p / HW;
    const int q = p - b * HW;
    const float* __restrict__ fp = feat + (size_t)b * (Cdim * HW) + q;

    const int klo    = (lane >> 4) << 1;         // 0 (lanes 0-15) or 2 (lanes 16-31)
    const int n_lo   = lane & 15;                // tile0 column (centroid 0..15)
    const int n_hi   = 16 + n_lo;                // tile1 column (centroid 16..31)
    const bool hi_ok = (n_hi < Kcen);

    const float* __restrict__ brow0 = &lds_cent[n_lo * LDS_STRIDE];
    const float* __restrict__ brow1 = &lds_cent[(hi_ok ? n_hi : 0) * LDS_STRIDE];

    v8f acc0 = {};   // dots vs centroids 0..15
    v8f acc1 = {};   // dots vs centroids 16..31 (only 16..18 valid)

    for (int c0 = 0; c0 < Cdim; c0 += 4) {
        const int cA = c0 + klo;
        v2f a;
        a.x = fp[(size_t)cA * HW];
        a.y = fp[(size_t)(cA + 1) * HW];
        v2f bv0 = *(const v2f*)(brow0 + cA);
        v2f bv1 = *(const v2f*)(brow1 + cA);
        acc0 = __builtin_amdgcn_wmma_f32_16x16x4_f32(
            false, a, false, bv0, (short)0, acc0, false, false);
        acc1 = __builtin_amdgcn_wmma_f32_16x16x4_f32(
            false, a, false, bv1, (short)0, acc1, false, false);
    }

    const float csq_lo = lds_csq[n_lo];
    const float csq_hi = lds_csq[hi_ok ? n_hi : 0];

    // C/D layout: VGPR r -> M=r (lanes 0-15) / M=r+8 (lanes 16-31), N = lane&15.
    // Argmin over k is a cross-lane min within each 16-lane half.
#pragma unroll
    for (int r = 0; r < 8; ++r) {
        float d  = csq_lo - 2.0f * acc0[r];
        int   bi = n_lo;
        if (hi_ok) {
            float d1 = csq_hi - 2.0f * acc1[r];
            if (d1 < d) { d = d1; bi = n_hi; }
        }
#pragma unroll
        for (int m = 1; m <= 8; m <<= 1) {
            float od = __shfl_xor(d, m, 32);
            int   ob = __shfl_xor(bi, m, 32);
            if (od < d || (od == d && ob < bi)) { d = od; bi = ob; }  // first-index tie-break
        }
        if ((lane & 15) == r) {
            const int M   = r + ((lane >> 4) << 3);
            const int pix = tile + M;
            if (pix < NPIX) cls_out[(size_t)t * NPIX + pix] = bi;
        }
    }
}

// Stage 2: nearest-neighbor upsample 65x129 -> 512x1024, emit class id as float.
// Integer src index math is exact (65/512, 129/1024 exactly representable; products < 2^24).
__global__ __launch_bounds__(256) void resize_kernel(
    const int* __restrict__ cls, float* __restrict__ out)
{
    const int t = blockIdx.y;
    const int o = blockIdx.x * 256 + threadIdx.x;
    if (o >= PERT) return;
    const int b   = o >> 19;              // 512*1024 = 2^19
    const int rem = o & ((1 << 19) - 1);
    const int Y   = rem >> 10;
    const int X   = rem & 1023;
    const int y   = (Y * Hh) >> 9;        // floor(Y * 65/512)
    const int x   = (X * Ww) >> 10;       // floor(X * 129/1024)
    const int src = b * HW + y * Ww + x;
    out[(size_t)t * PERT + o] = (float)cls[(size_t)t * NPIX + src];
}

extern "C" void kernel_launch(void* const* d_in, const int* in_sizes, int n_in,
                              void* d_out, int out_size, void* d_ws, size_t ws_size,
                              hipStream_t stream) {
    const float* f_s2t = (const float*)d_in[0];
    const float* f_tgt = (const float*)d_in[1];
    const float* c_s2t = (const float*)d_in[2];
    const float* c_tgt = (const float*)d_in[3];
    // seg_s2t / seg_target only contribute their (fixed) shapes.

    int* cls = (int*)d_ws;  // 2 * NPIX int32 = 536,640 bytes of scratch

    dim3 g1((NPIX + PIX_PER_BLOCK - 1) / PIX_PER_BLOCK, 2);
    cls_argmin_kernel<<<g1, 256, 0, stream>>>(f_s2t, f_tgt, c_s2t, c_tgt, cls);

    dim3 g2((PERT + 255) / 256, 2);
    resize_kernel<<<g2, 256, 0, stream>>>(cls, (float*)d_out);
}